// Autoencoder_layers_byhalf_SPDnet_2619930050823
// MI455X (gfx1250) — compile-verified
//
#include <hip/hip_runtime.h>

// ---------------------------------------------------------------------------
// SPDNet autoencoder, fully fused per batch element on gfx1250 (MI455X).
//
// ReEig / spectral ops are computed with Newton-Schulz matrix-sign iterations
// (GEMM-only -> v_wmma_f32_16x16x4_f32), not eigensolvers:
//   M = H - eps*I ;  S0 = M/||M||_F ;  S <- 1.5 S - 0.5 S^3 (NS_ITERS times)
//   |M| ~= S*M ;  ReEig(H) = 0.5*M + 0.5*sym(S*M) + eps*I
// LogEig followed by ExpEig is an exact identity and is elided.
// Everything for one batch element stays resident in LDS (~295 KB < 320 KB).
// ---------------------------------------------------------------------------

typedef __attribute__((ext_vector_type(2))) float v2f;
typedef __attribute__((ext_vector_type(8))) float v8f;

#define WAVES    8
#define TPB      (WAVES * 32)
#define SPD_EPS  1e-4f
#define NS_ITERS 24          // 1.5x growth/iter -> resolves |lambda|/||M|| >= ~1e-5

// C(MxN) = A(MxK) * op(B),  op(B)=B (KxN, ldb) or B^T with B stored (NxK, ldb)
// All operands in LDS, row-major. M,N,K multiples of 16 (here 16/32/64/128).
__device__ __forceinline__ void gemm_wmma(const float* __restrict__ A, int lda,
                                          const float* __restrict__ B, int ldb,
                                          float* __restrict__ C, int ldc,
                                          int M, int N, int K, bool transB)
{
    const int lane = threadIdx.x & 31;
    const int wave = threadIdx.x >> 5;
    const int half = lane >> 4;      // 0: lanes 0-15, 1: lanes 16-31
    const int l16  = lane & 15;
    const int tm = M >> 4, tn = N >> 4;

    for (int t = wave; t < tm * tn; t += WAVES) {
        const int ti = (t / tn) << 4;
        const int tj = (t % tn) << 4;
        v8f c = {0.f, 0.f, 0.f, 0.f, 0.f, 0.f, 0.f, 0.f};

        for (int kk = 0; kk < K; kk += 4) {
            // A fragment 16x4: VGPR0 holds K=2*half, VGPR1 holds K=2*half+1
            v2f a;
            a[0] = A[(ti + l16) * lda + kk + 2 * half + 0];
            a[1] = A[(ti + l16) * lda + kk + 2 * half + 1];
            // B fragment 4x16: row K=2*half(+1), columns tj..tj+15
            v2f bf;
            if (!transB) {
                bf[0] = B[(kk + 2 * half + 0) * ldb + tj + l16];
                bf[1] = B[(kk + 2 * half + 1) * ldb + tj + l16];
            } else {  // op(B)(k,j) = B[j*ldb + k]
                bf[0] = B[(tj + l16) * ldb + kk + 2 * half + 0];
                bf[1] = B[(tj + l16) * ldb + kk + 2 * half + 1];
            }
            c = __builtin_amdgcn_wmma_f32_16x16x4_f32(
                    false, a, false, bf, (short)0, c, false, false);
        }
        // C/D layout: VGPR r -> row ti + r + 8*half, col tj + l16
#pragma unroll
        for (int r = 0; r < 8; ++r)
            C[(ti + r + 8 * half) * ldc + tj + l16] = c[r];
    }
}

// ReEig on h (s x s).  Mb, S, T are s*s scratch; red is 1-float LDS scalar.
__device__ __forceinline__ void reeig(float* h, float* Mb, float* S, float* T,
                                      float* red, int s)
{
    const int n2 = s * s;
    // Mb = h - eps*I ; accumulate ||Mb||_F^2
    if (threadIdx.x == 0) *red = 0.f;
    __syncthreads();
    float part = 0.f;
    for (int i = threadIdx.x; i < n2; i += TPB) {
        int r = i / s, c = i % s;
        float v = h[i] - (r == c ? SPD_EPS : 0.f);
        Mb[i] = v;
        part += v * v;
    }
    atomicAdd(red, part);            // ds_add_f32
    __syncthreads();
    const float inv = rsqrtf(*red + 1e-30f);
    for (int i = threadIdx.x; i < n2; i += TPB) S[i] = Mb[i] * inv;
    __syncthreads();

    // Newton-Schulz sign iteration: S <- 1.5 S - 0.5 S^3   (h reused as temp)
    for (int it = 0; it < NS_ITERS; ++it) {
        gemm_wmma(S, s, S, s, T, s, s, s, s, false);   // T = S*S
        __syncthreads();
        gemm_wmma(T, s, S, s, h, s, s, s, s, false);   // h = S^3
        __syncthreads();
        for (int i = threadIdx.x; i < n2; i += TPB)
            S[i] = 1.5f * S[i] - 0.5f * h[i];
        __syncthreads();
    }

    gemm_wmma(S, s, Mb, s, T, s, s, s, s, false);      // T = sign(M)*M = |M|
    __syncthreads();
    for (int i = threadIdx.x; i < n2; i += TPB) {
        int r = i / s, c = i % s;
        h[i] = 0.5f * Mb[i] + 0.25f * (T[i] + T[c * s + r])   // symmetrized
             + (r == c ? SPD_EPS : 0.f);
    }
    __syncthreads();
}

// h (n x n) -> h = W h W^T (m x m); W global, (m x n) row-major.
__device__ __forceinline__ void bimap(float* h, float* Wl, float* T1,
                                      const float* __restrict__ Wg,
                                      int m, int n)
{
    for (int i = threadIdx.x; i < m * n; i += TPB) Wl[i] = Wg[i];
    __syncthreads();
    gemm_wmma(Wl, n, h, n, T1, n, m, n, n, false);     // T1 = W h     (m x n)
    __syncthreads();
    gemm_wmma(T1, n, Wl, n, h, m, m, m, n, true);      // h  = T1 W^T  (m x m)
    __syncthreads();
}

__global__ __launch_bounds__(TPB)
void Autoencoder_layers_byhalf_SPDnet_2619930050823_kernel(
        const float* __restrict__ X,
        const float* __restrict__ we0, const float* __restrict__ we1,
        const float* __restrict__ we2,
        const float* __restrict__ wd0, const float* __restrict__ wd1,
        const float* __restrict__ wd2,
        float* __restrict__ out)
{
    extern __shared__ float smem[];
    float* h   = smem;                         // 128*128
    float* Mb  = smem + 16384;                 // 128*128
    float* T1  = smem + 2 * 16384;             // 128*128
    float* T2  = smem + 3 * 16384;             // 128*128
    float* Wl  = smem + 4 * 16384;             // up to 128*64
    float* red = smem + 4 * 16384 + 8192;      // scalar

    const int b = blockIdx.x;
    const float* Xb = X + (size_t)b * 16384;
    for (int i = threadIdx.x; i < 16384; i += TPB) h[i] = Xb[i];
    __syncthreads();

    // ---- encoder: BiMap + ReEig, 128 -> 64 -> 32 -> 16
    bimap(h, Wl, T1, we0, 64, 128);  reeig(h, Mb, T1, T2, red, 64);
    bimap(h, Wl, T1, we1, 32, 64);   reeig(h, Mb, T1, T2, red, 32);
    bimap(h, Wl, T1, we2, 16, 32);   reeig(h, Mb, T1, T2, red, 16);

    // ---- LogEig -> ExpEig is an exact identity; head reduces to ReEig(16)
    reeig(h, Mb, T1, T2, red, 16);

    // ---- decoder: BiMap + ReEig, 16 -> 32 -> 64 -> 128
    bimap(h, Wl, T1, wd0, 32, 16);   reeig(h, Mb, T1, T2, red, 32);
    bimap(h, Wl, T1, wd1, 64, 32);   reeig(h, Mb, T1, T2, red, 64);
    bimap(h, Wl, T1, wd2, 128, 64);  reeig(h, Mb, T1, T2, red, 128);

    float* Ob = out + (size_t)b * 16384;
    for (int i = threadIdx.x; i < 16384; i += TPB) Ob[i] = h[i];
}

extern "C" void kernel_launch(void* const* d_in, const int* in_sizes, int n_in,
                              void* d_out, int out_size, void* d_ws, size_t ws_size,
                              hipStream_t stream)
{
    const float* x   = (const float*)d_in[0];
    const float* we0 = (const float*)d_in[1];
    const float* we1 = (const float*)d_in[2];
    const float* we2 = (const float*)d_in[3];
    const float* wd0 = (const float*)d_in[4];
    const float* wd1 = (const float*)d_in[5];
    const float* wd2 = (const float*)d_in[6];
    float* out = (float*)d_out;

    const int batch = in_sizes[0] / (128 * 128);
    const size_t smem_bytes = (size_t)(4 * 16384 + 8192 + 32) * sizeof(float);

    (void)hipFuncSetAttribute(
        reinterpret_cast<const void*>(
            &Autoencoder_layers_byhalf_SPDnet_2619930050823_kernel),
        hipFuncAttributeMaxDynamicSharedMemorySize, (int)smem_bytes);

    Autoencoder_layers_byhalf_SPDnet_2619930050823_kernel
        <<<batch, TPB, smem_bytes, stream>>>(x, we0, we1, we2, wd0, wd1, wd2, out);
}